// DraftModel_2997887172795
// MI455X (gfx1250) — compile-verified
//
#include <hip/hip_runtime.h>
#include <hip/hip_bf16.h>
#include <math.h>

#define D_MODEL 4096
#define VOCAB   32000
#define TOPK    10
#define DEPTH   5
#define NTOK    59          // total_tokens
#define DMAX    (DEPTH + 2) // 7

#define KSPLIT_H 4          // head GEMM K-split (slices of 1024)
#define KSPLIT_F 8          // fc GEMM K-split (slices of 1024)
#define NBLK_H   250        // column-tile blocks per K-slice (250*8*16 = 32000)
#define NBLK_F   32         // column-tile blocks per K-slice (32*8*16 = 4096)

typedef __attribute__((ext_vector_type(2))) float v2f;
typedef __attribute__((ext_vector_type(8))) float v8f;

// ---------------------------------------------------------------------------
// WMMA f32 16x16x4:  D(16x16,f32) = A(16x4,f32) x B(4x16,f32) + C
// Lane mapping (wave32):
//   A frag (v2f): lane l -> M = l%16, element j -> K = j + 2*(l/16)
//   B frag (v2f): lane l -> N = l%16, element j -> K = j + 2*(l/16)
//   C/D (v8f):    element m -> M = m + 8*(l/16), N = l%16
// ---------------------------------------------------------------------------
__device__ __forceinline__ v8f wmma4(v2f a, v2f b, v8f c) {
  return __builtin_amdgcn_wmma_f32_16x16x4_f32(false, a, false, b, (short)0, c,
                                               false, false);
}

// --------------------------- padded hidden (16x4096) -----------------------
__global__ void __launch_bounds__(256)
k_pad_hidden(const float* __restrict__ lh, float* __restrict__ H0) {
  int i = blockIdx.x * 256 + threadIdx.x;           // grid = 256 -> 65536 elems
  if (i < 16 * D_MODEL) H0[i] = (i < D_MODEL) ? lh[i] : 0.f;
}

// ---------- head GEMM partials: part[ks](16x32000) = H[:,ks] x Wh[ks] ------
// grid: KSPLIT_H * NBLK_H blocks x 256 threads (8 waves), 16 cols per wave.
// A staged in LDS in 512-wide K chunks; W_head streamed with NT hint.
__global__ void __launch_bounds__(256)
k_head_gemm(const float* __restrict__ H, const float* __restrict__ Wh,
            float* __restrict__ part) {
  __shared__ __align__(16) float sA[16 * 512];
  const int tid = threadIdx.x;
  const int lane = tid & 31, wid = tid >> 5;
  const int lo = lane & 15, hi = lane >> 4;
  const int ks = blockIdx.x / NBLK_H;               // K-slice
  const int bt = blockIdx.x % NBLK_H;
  const int n0 = (bt * 8 + wid) * 16;               // vocab tile base
  const int k0 = ks * (D_MODEL / KSPLIT_H);         // 1024-wide slice
  v8f c = {};
  for (int kc = k0; kc < k0 + D_MODEL / KSPLIT_H; kc += 512) {
    __syncthreads();
    for (int q = tid; q < 16 * 128; q += 256) {     // 16x512 floats as float4
      const int row = q >> 7, col4 = q & 127;
      ((float4*)sA)[q] = *(const float4*)(H + row * D_MODEL + kc + col4 * 4);
    }
    __syncthreads();
#pragma unroll 8
    for (int k = 0; k < 512; k += 4) {
      v2f a = *(const v2f*)&sA[lo * 512 + k + 2 * hi];
      const float* wp = Wh + (size_t)(kc + k + 2 * hi) * VOCAB + n0 + lo;
      v2f b;
      b.x = __builtin_nontemporal_load(wp);          // stream: don't pollute L2
      b.y = __builtin_nontemporal_load(wp + VOCAB);
      c = wmma4(a, b, c);
    }
  }
  float* out = part + (size_t)ks * 16 * VOCAB;
#pragma unroll
  for (int m = 0; m < 8; ++m)
    out[(size_t)(m + 8 * hi) * VOCAB + n0 + lo] = c[m];
}

// ------------- fixed-order reduction of head partials (deterministic) ------
__global__ void __launch_bounds__(256)
k_reduce_head(const float* __restrict__ part, float* __restrict__ logits) {
  const int i = blockIdx.x * 256 + threadIdx.x;      // float4 idx, 128000 total
  const float4* p = (const float4*)part;
  float4 s = p[i];
#pragma unroll
  for (int k = 1; k < KSPLIT_H; ++k) {
    float4 q = p[(size_t)k * (16 * VOCAB / 4) + i];
    s.x += q.x; s.y += q.y; s.z += q.z; s.w += q.w;
  }
  ((float4*)logits)[i] = s;
}

// ---------- fc GEMM partials: part[ks](16x4096) = X[:,ks] x Wfc[ks] --------
// grid: KSPLIT_F * NBLK_F blocks; W_fc (134MB) left cacheable in 192MB L2.
__global__ void __launch_bounds__(256)
k_fc_gemm(const float* __restrict__ X, const float* __restrict__ Wfc,
          float* __restrict__ part) {
  __shared__ __align__(16) float sA[16 * 512];
  const int tid = threadIdx.x;
  const int lane = tid & 31, wid = tid >> 5;
  const int lo = lane & 15, hi = lane >> 4;
  const int ks = blockIdx.x / NBLK_F;
  const int bt = blockIdx.x % NBLK_F;
  const int n0 = (bt * 8 + wid) * 16;               // output-channel tile
  const int k0 = ks * (2 * D_MODEL / KSPLIT_F);     // 1024-wide slice
  v8f c = {};
  for (int kc = k0; kc < k0 + 2 * D_MODEL / KSPLIT_F; kc += 512) {
    __syncthreads();
    for (int q = tid; q < 16 * 128; q += 256) {
      const int row = q >> 7, col4 = q & 127;
      ((float4*)sA)[q] = *(const float4*)(X + row * (2 * D_MODEL) + kc + col4 * 4);
    }
    __syncthreads();
#pragma unroll 8
    for (int k = 0; k < 512; k += 4) {
      v2f a = *(const v2f*)&sA[lo * 512 + k + 2 * hi];
      const float* wp = Wfc + (size_t)(kc + k + 2 * hi) * D_MODEL + n0 + lo;
      v2f b;
      b.x = wp[0];
      b.y = wp[D_MODEL];
      c = wmma4(a, b, c);
    }
  }
  float* out = part + (size_t)ks * 16 * D_MODEL;
#pragma unroll
  for (int m = 0; m < 8; ++m)
    out[(size_t)(m + 8 * hi) * D_MODEL + n0 + lo] = c[m];
}

// ---------- fixed-order reduction of fc partials + tanh activation ---------
__global__ void __launch_bounds__(256)
k_reduce_fc_tanh(const float* __restrict__ part, float* __restrict__ outH) {
  const int i = blockIdx.x * 256 + threadIdx.x;      // float4 idx, 16384 total
  const float4* p = (const float4*)part;
  float4 s = p[i];
#pragma unroll
  for (int k = 1; k < KSPLIT_F; ++k) {
    float4 q = p[(size_t)k * (16 * D_MODEL / 4) + i];
    s.x += q.x; s.y += q.y; s.z += q.z; s.w += q.w;
  }
  float4 r;
  r.x = tanhf(s.x); r.y = tanhf(s.y); r.z = tanhf(s.z); r.w = tanhf(s.w);
  ((float4*)outH)[i] = r;
}

// --------------- per-row max + logsumexp (for log_softmax) -----------------
__global__ void __launch_bounds__(256)
k_rowstats(const float* __restrict__ logits, float* __restrict__ rowoff) {
  __shared__ float red[256];
  const int r = blockIdx.x, tid = threadIdx.x;
  const float* row = logits + (size_t)r * VOCAB;
  float mx = -INFINITY;
  for (int v = tid; v < VOCAB; v += 256) mx = fmaxf(mx, row[v]);
  red[tid] = mx; __syncthreads();
  for (int s = 128; s > 0; s >>= 1) {
    if (tid < s) red[tid] = fmaxf(red[tid], red[tid + s]);
    __syncthreads();
  }
  mx = red[0]; __syncthreads();
  float sum = 0.f;
  for (int v = tid; v < VOCAB; v += 256) sum += expf(row[v] - mx);
  red[tid] = sum; __syncthreads();
  for (int s = 128; s > 0; s >>= 1) {
    if (tid < s) red[tid] += red[tid + s];
    __syncthreads();
  }
  if (tid == 0) rowoff[r] = mx + logf(red[0]);
}

// ------------- per-row top-10 (descending, ties -> lowest index) -----------
__global__ void __launch_bounds__(256)
k_topk_row(const float* __restrict__ logits, const float* __restrict__ rowoff,
           float* __restrict__ tv, int* __restrict__ tidx) {
  __shared__ float rv[256]; __shared__ int ri[256];
  __shared__ int picked[TOPK]; __shared__ float pval[TOPK];
  const int r = blockIdx.x, tid = threadIdx.x;
  const float* row = logits + (size_t)r * VOCAB;
  for (int it = 0; it < TOPK; ++it) {
    float bv = -INFINITY; int bi = VOCAB;
    for (int v = tid; v < VOCAB; v += 256) {
      bool skip = false;
      for (int t = 0; t < it; ++t) skip |= (picked[t] == v);
      if (!skip) { float x = row[v]; if (x > bv) { bv = x; bi = v; } }
    }
    rv[tid] = bv; ri[tid] = bi; __syncthreads();
    for (int s = 128; s > 0; s >>= 1) {
      if (tid < s) {
        float ov = rv[tid + s]; int oi = ri[tid + s];
        if (ov > rv[tid] || (ov == rv[tid] && oi < ri[tid])) { rv[tid] = ov; ri[tid] = oi; }
      }
      __syncthreads();
    }
    if (tid == 0) { picked[it] = ri[0]; pval[it] = rv[0]; }
    __syncthreads();
  }
  if (tid < TOPK) {
    tv[r * TOPK + tid] = pval[tid] - rowoff[r];   // log_softmax value
    tidx[r * TOPK + tid] = picked[tid];
  }
}

// ------------------------------ level 0 init -------------------------------
__global__ void k_init_level(const float* __restrict__ tv, const int* __restrict__ tidx,
                             float* __restrict__ scores, float* __restrict__ scores_flat,
                             int* __restrict__ ss_flat, int* __restrict__ parents_flat,
                             int* __restrict__ input_ids, int* __restrict__ rowmap,
                             int* __restrict__ cs_idx) {
  int t = threadIdx.x;
  if (t < TOPK) {
    scores[t] = tv[t]; scores_flat[t] = tv[t];
    ss_flat[t] = tidx[t]; input_ids[t] = tidx[t];
    rowmap[t] = 0; cs_idx[t] = t;
  }
  if (t == 0) parents_flat[0] = 0;
}

// ---------------- X = [ E[input_ids[r]] , srcH[rowmap[r]] ] ----------------
__global__ void __launch_bounds__(256)
k_gather_concat(float* __restrict__ X, const float* __restrict__ E,
                const float* __restrict__ srcH,
                const int* __restrict__ input_ids, const int* __restrict__ rowmap) {
  const int r = blockIdx.x, tid = threadIdx.x;
  float* xr = X + (size_t)r * (2 * D_MODEL);
  if (r < TOPK) {
    const float* e = E + (size_t)input_ids[r] * D_MODEL;
    const float* h = srcH + (size_t)rowmap[r] * D_MODEL;
    for (int c = tid; c < D_MODEL; c += 256) { xr[c] = e[c]; xr[D_MODEL + c] = h[c]; }
  } else {
    for (int c = tid; c < 2 * D_MODEL; c += 256) xr[c] = 0.f;
  }
}

// -------- cumulative scores, flat top-10, parents, gather indices ----------
__global__ void __launch_bounds__(128)
k_level_combine(int level, const float* __restrict__ tv, const int* __restrict__ tidx,
                float* __restrict__ scores, float* __restrict__ scores_flat,
                int* __restrict__ ss_flat, int* __restrict__ parents_flat,
                int* __restrict__ input_ids, int* __restrict__ rowmap,
                int* __restrict__ cs_idx) {
  __shared__ float cu[TOPK * TOPK];
  const int t = threadIdx.x;
  const int bias1 = (level > 0) ? TOPK : 0;
  const int bias2 = (level > 1) ? (level - 1) : 0;
  const int bias = 1 + TOPK * TOPK * bias2 + bias1;
  if (t < TOPK) parents_flat[1 + level * TOPK + t] = cs_idx[t] + bias;
  __syncthreads();
  if (t < TOPK * TOPK) {
    float v = tv[t] + scores[t / TOPK];
    cu[t] = v;
    scores_flat[TOPK + level * 100 + t] = v;
    ss_flat[TOPK + level * 100 + t] = tidx[t];
  }
  __syncthreads();
  if (t == 0) {
    for (int it = 0; it < TOPK; ++it) {
      float bv = -INFINITY; int bi = 0;
      for (int j = 0; j < TOPK * TOPK; ++j)
        if (cu[j] > bv) { bv = cu[j]; bi = j; }
      cu[bi] = -INFINITY;
      scores[it] = bv; cs_idx[it] = bi;
      rowmap[it] = bi / TOPK;
      input_ids[it] = tidx[bi];
    }
  }
}

// --------------------------- final tree build ------------------------------
__global__ void __launch_bounds__(64)
k_build_tree(const float* __restrict__ scores_flat, const int* __restrict__ ss_flat,
             const int* __restrict__ parents_flat, const int* __restrict__ sample_token,
             float* __restrict__ out) {
  __shared__ float sc[510];
  __shared__ int tsi[NTOK], mi[NTOK], p[NTOK + 1], anc[(NTOK + 1) * DMAX], pos[NTOK + 1];
  __shared__ unsigned char noleaf[NTOK + 1];
  float* o_tok  = out;                 // 60
  float* o_ret  = out + 60;            // 60*7 = 420
  float* o_mask = out + 480;           // 60*60 = 3600
  float* o_pos  = out + 4080;          // 60
  const int tid = threadIdx.x;
  for (int q = tid; q < 510; q += 64) sc[q] = scores_flat[q];
  for (int q = tid; q < (NTOK + 1) * (NTOK + 1); q += 64) o_mask[q] = 0.f;
  for (int q = tid; q <= NTOK; q += 64) noleaf[q] = 0;
  __syncthreads();
  if (tid == 0) {
    // top-59 (descending, ties -> lowest index), then sort ascending
    for (int it = 0; it < NTOK; ++it) {
      float bv = -INFINITY; int bi = 0;
      for (int j = 0; j < 510; ++j) if (sc[j] > bv) { bv = sc[j]; bi = j; }
      sc[bi] = -INFINITY; tsi[it] = bi;
    }
    for (int i = 1; i < NTOK; ++i) {
      int key = tsi[i], j = i - 1;
      while (j >= 0 && tsi[j] > key) { tsi[j + 1] = tsi[j]; --j; }
      tsi[j + 1] = key;
    }
    o_tok[0] = (float)sample_token[0];
    for (int t = 0; t < NTOK; ++t) o_tok[1 + t] = (float)ss_flat[tsi[t]];
    for (int t = 0; t < NTOK; ++t) {
      int dp = parents_flat[tsi[t] / TOPK];
      if (dp == 0) mi[t] = 0;
      else {
        int v = dp - 1, ssi = NTOK;
        for (int j = 0; j < NTOK; ++j) if (tsi[j] >= v) { ssi = j; break; }
        mi[t] = ssi + 1;
      }
    }
    p[0] = 0;
    for (int t = 0; t < NTOK; ++t) p[t + 1] = mi[t];
    for (int i = 0; i <= NTOK; ++i) {
      anc[i * DMAX] = i;
      for (int j = 1; j < DMAX; ++j) anc[i * DMAX + j] = p[anc[i * DMAX + j - 1]];
    }
    for (int i = 0; i <= NTOK; ++i) {
      for (int j = 0; j < DMAX; ++j) o_mask[i * (NTOK + 1) + anc[i * DMAX + j]] = 1.f;
      o_mask[i * (NTOK + 1)] = 1.f;
    }
    for (int i = 0; i <= NTOK; ++i) {
      int s = 0;
      for (int j = 0; j <= NTOK; ++j) s += (o_mask[i * (NTOK + 1) + j] != 0.f);
      pos[i] = s - 1; o_pos[i] = (float)pos[i];
    }
    for (int t = 0; t < NTOK; ++t) noleaf[mi[t]] = 1;
    for (int i = 0; i <= NTOK; ++i) {
      for (int c = 0; c < DMAX; ++c) {
        int kk = pos[i] - c;
        int kkc = kk < 0 ? 0 : (kk > DMAX - 1 ? DMAX - 1 : kk);
        float val = (kk >= 0) ? (float)anc[i * DMAX + kkc] : -1.f;
        o_ret[i * DMAX + c] = noleaf[i] ? -1.f : val;
      }
    }
  }
}

// ===========================================================================
extern "C" void kernel_launch(void* const* d_in, const int* in_sizes, int n_in,
                              void* d_out, int out_size, void* d_ws, size_t ws_size,
                              hipStream_t stream) {
  const float* last_hidden = (const float*)d_in[0];
  const float* E           = (const float*)d_in[1];
  const float* Wfc         = (const float*)d_in[2];
  const float* Whead       = (const float*)d_in[3];
  const int*   sample_tok  = (const int*)d_in[4];

  char* ws = (char*)d_ws;
  size_t off = 0;
  auto alloc = [&](size_t bytes) -> char* {
    char* pp = ws + off;
    off = (off + bytes + 255) & ~(size_t)255;
    return pp;
  };
  float* H0          = (float*)alloc(16 * D_MODEL * 4);
  float* X           = (float*)alloc(16 * 2 * D_MODEL * 4);
  float* outH        = (float*)alloc(16 * D_MODEL * 4);
  float* logits      = (float*)alloc((size_t)16 * VOCAB * 4);
  float* head_part   = (float*)alloc((size_t)KSPLIT_H * 16 * VOCAB * 4);
  float* fc_part     = (float*)alloc((size_t)KSPLIT_F * 16 * D_MODEL * 4);
  float* rowoff      = (float*)alloc(16 * 4);
  float* tv          = (float*)alloc(100 * 4);
  int*   tidx        = (int*)alloc(100 * 4);
  float* scores      = (float*)alloc(16 * 4);
  float* scores_flat = (float*)alloc(512 * 4);
  int*   ss_flat     = (int*)alloc(512 * 4);
  int*   parents     = (int*)alloc(64 * 4);
  int*   input_ids   = (int*)alloc(16 * 4);
  int*   rowmap      = (int*)alloc(16 * 4);
  int*   cs_idx      = (int*)alloc(16 * 4);
  if (off > ws_size) return;  // workspace too small; nothing safe to do

  // level 0: logits over last_hidden, top-10
  k_pad_hidden<<<256, 256, 0, stream>>>(last_hidden, H0);
  k_head_gemm<<<KSPLIT_H * NBLK_H, 256, 0, stream>>>(H0, Whead, head_part);
  k_reduce_head<<<500, 256, 0, stream>>>(head_part, logits);
  k_rowstats<<<1, 256, 0, stream>>>(logits, rowoff);
  k_topk_row<<<1, 256, 0, stream>>>(logits, rowoff, tv, tidx);
  k_init_level<<<1, 32, 0, stream>>>(tv, tidx, scores, scores_flat, ss_flat,
                                     parents, input_ids, rowmap, cs_idx);

  for (int i = 0; i < DEPTH; ++i) {
    const float* srcH = (i == 0) ? H0 : outH;
    k_gather_concat<<<16, 256, 0, stream>>>(X, E, srcH, input_ids, rowmap);
    k_fc_gemm<<<KSPLIT_F * NBLK_F, 256, 0, stream>>>(X, Wfc, fc_part);
    k_reduce_fc_tanh<<<64, 256, 0, stream>>>(fc_part, outH);
    k_head_gemm<<<KSPLIT_H * NBLK_H, 256, 0, stream>>>(outH, Whead, head_part);
    k_reduce_head<<<500, 256, 0, stream>>>(head_part, logits);
    k_rowstats<<<TOPK, 256, 0, stream>>>(logits, rowoff);
    k_topk_row<<<TOPK, 256, 0, stream>>>(logits, rowoff, tv, tidx);
    k_level_combine<<<1, 128, 0, stream>>>(i, tv, tidx, scores, scores_flat,
                                           ss_flat, parents, input_ids, rowmap,
                                           cs_idx);
  }
  k_build_tree<<<1, 64, 0, stream>>>(scores_flat, ss_flat, parents, sample_tok,
                                     (float*)d_out);
}